// blockLayer_48653389529228
// MI455X (gfx1250) — compile-verified
//
#include <hip/hip_runtime.h>

typedef __attribute__((ext_vector_type(2))) float v2f;
typedef __attribute__((ext_vector_type(8))) float v8f;

#define BATCH 4096
#define NBLK  8192
#define COLS  (NBLK * 3)        // 24576
#define ROW_TILES (BATCH / 16)  // 256
#define COL_TILES (COLS / 16)   // 1536
#define CT_PER_WAVE 4
#define WAVES_PER_BLOCK 8

__device__ __forceinline__ float softplus_f(float x) {
    // numerically stable softplus, matches jax.nn.softplus
    return (x > 20.0f) ? x : log1pf(__expf(x));
}

// ---------------------------------------------------------------------------
// Prep: build W[d, n*3+k] = B[n][d][k] with B = M^T M, M upper-triangular 3x3.
// W is padded to 4 rows x 24576 cols (row 3 = zeros) so the GEMM's K=4 WMMA
// can load all B slices unconditionally -> no EXEC-divergent guard branches.
// ---------------------------------------------------------------------------
__global__ void blockLayer_prep_kernel(const float* __restrict__ w,
                                       float* __restrict__ W) {
    int n = blockIdx.x * blockDim.x + threadIdx.x;
    if (n >= NBLK) return;
    const float* wn = w + n * 6;
    float a = softplus_f(wn[0]);  // M00
    float b = wn[1];              // M01
    float c = wn[2];              // M02
    float d = softplus_f(wn[3]);  // M11
    float e = wn[4];              // M12
    float f = softplus_f(wn[5]);  // M22
    // B = M^T M (symmetric)
    float B00 = a * a;
    float B01 = a * b;
    float B02 = a * c;
    float B11 = b * b + d * d;
    float B12 = b * c + d * e;
    float B22 = c * c + e * e + f * f;
    size_t base = (size_t)n * 3;
    W[0 * (size_t)COLS + base + 0] = B00;
    W[0 * (size_t)COLS + base + 1] = B01;
    W[0 * (size_t)COLS + base + 2] = B02;
    W[1 * (size_t)COLS + base + 0] = B01;
    W[1 * (size_t)COLS + base + 1] = B11;
    W[1 * (size_t)COLS + base + 2] = B12;
    W[2 * (size_t)COLS + base + 0] = B02;
    W[2 * (size_t)COLS + base + 1] = B12;
    W[2 * (size_t)COLS + base + 2] = B22;
    // K=3 padding row: zeros
    W[3 * (size_t)COLS + base + 0] = 0.0f;
    W[3 * (size_t)COLS + base + 1] = 0.0f;
    W[3 * (size_t)COLS + base + 2] = 0.0f;
}

// ---------------------------------------------------------------------------
// GEMM: out(4096 x 24576) = input(4096 x 3, padded to K=4) @ W(4 x 24576).
// One wave per 16x16 tile (x CT_PER_WAVE col tiles), V_WMMA_F32_16X16X4_F32.
// ---------------------------------------------------------------------------
__global__ void __launch_bounds__(32 * WAVES_PER_BLOCK)
blockLayer_gemm_kernel(const float* __restrict__ inp,
                       const float* __restrict__ W,
                       float* __restrict__ out) {
    const int lane = threadIdx.x & 31;
    const int wave = threadIdx.x >> 5;
    const int gwave = blockIdx.x * WAVES_PER_BLOCK + wave;

    const int col_groups = COL_TILES / CT_PER_WAVE;  // 384
    const int row_tile  = gwave / col_groups;        // 0..255
    const int col_group = gwave % col_groups;        // 0..383

    const int m  = lane & 15;   // row (A) / col (B, D) within tile
    const int hi = lane >> 4;   // half-wave select
    const int k0 = hi * 2;      // K index for VGPR0 of A/B: 0 or 2

    // --- A tile: 16x4 f32, lanes 0-15 hold K=0,1; lanes 16-31 hold K=2,3 ---
    // Unconditional loads + select (no EXEC divergence): K=3 slice is zero.
    const int arow = row_tile * 16 + m;
    const float* ar = inp + (size_t)arow * 3;
    float ax = ar[k0];        // k0 in {0,2}: always valid
    float ay = ar[1];         // always in-bounds; only used when hi==0
    v2f a;
    a.x = ax;
    a.y = hi ? 0.0f : ay;     // v_cndmask, EXEC untouched

    const int col_base = col_group * (16 * CT_PER_WAVE);
    float* orow = out + (size_t)(row_tile * 16 + hi * 8) * COLS;

#pragma unroll
    for (int t = 0; t < CT_PER_WAVE; ++t) {
        const int col = col_base + t * 16 + m;

        // --- B tile: 4x16 f32, rows k0 and k0+1 (row 3 is the zero pad) ---
        v2f b;
        b.x = W[(size_t)k0 * COLS + col];
        b.y = W[(size_t)(k0 + 1) * COLS + col];

        v8f c = {};
        // D = A x B + 0  ->  v_wmma_f32_16x16x4_f32
        c = __builtin_amdgcn_wmma_f32_16x16x4_f32(
                /*neg_a=*/false, a, /*neg_b=*/false, b,
                /*c_mod=*/(short)0, c, /*reuse_a=*/false, /*reuse_b=*/false);

        // D layout: VGPR j holds M = j + hi*8, N = col. 8 coalesced b32
        // non-temporal stores (output is write-once; keep L2 for W/input).
#pragma unroll
        for (int j = 0; j < 8; ++j) {
            __builtin_nontemporal_store(c[j], &orow[(size_t)j * COLS + col]);
        }
    }
}

extern "C" void kernel_launch(void* const* d_in, const int* in_sizes, int n_in,
                              void* d_out, int out_size, void* d_ws, size_t ws_size,
                              hipStream_t stream) {
    const float* inp    = (const float*)d_in[0];  // (4096, 3)
    const float* weight = (const float*)d_in[1];  // (8192, 6)
    float*       out    = (float*)d_out;          // (4096, 24576)
    float*       W      = (float*)d_ws;           // 4 x 24576 f32 = 393 KB

    // Stage 1: materialize W = per-block (M^T M) as a K-padded GEMM B-matrix.
    blockLayer_prep_kernel<<<(NBLK + 255) / 256, 256, 0, stream>>>(weight, W);

    // Stage 2: store-bandwidth-bound WMMA GEMM producing the 402 MB output.
    const int total_waves = ROW_TILES * (COL_TILES / CT_PER_WAVE);  // 98304
    const int blocks = total_waves / WAVES_PER_BLOCK;               // 12288
    blockLayer_gemm_kernel<<<blocks, 32 * WAVES_PER_BLOCK, 0, stream>>>(inp, W, out);
}